// Word2Vec_79448305041435
// MI455X (gfx1250) — compile-verified
//
#include <hip/hip_runtime.h>
#include <hip/hip_bf16.h>
#include <math.h>

// Problem constants (match reference)
#define ORIG    19986
#define NEWROWS 37964
#define FACT    1024
#define DIM     768
#define BATCH   4096
#define KNEG    5

// Tiling for the fused softmax-GEMM (flash-attention style)
#define BM      16      // NEW rows per workgroup
#define BN      128     // vocab (ORIG) columns per chunk  (8 waves x 16)
#define KF      64      // f-panel depth staged in LDS
#define NPANEL  (FACT / KF)     // 16
#define NWAVES  8
#define QPITCH  1028    // 1024 + 4 pad  (bank-conflict-free A-frag reads)
#define PNPITCH 66      // 64 + 2 pad    (transposed K panel, b64-aligned)
#define PPITCH  132     // 128 + 4 pad   (P tile)

#define NCHUNK  ((ORIG + BN - 1) / BN)          // 157
#define GRID1   ((NEWROWS + BM - 1) / BM)       // 2373

// LDS layout (floats) — double-buffered K panel
#define LDS_Q   0
#define LDS_PNA (BM * QPITCH)                       // 16448
#define LDS_PNB (LDS_PNA + BN * PNPITCH)            // +8448
#define LDS_PB  (LDS_PNB + BN * PNPITCH)            // +8448
#define LDS_WM  (LDS_PB + BM * PPITCH)              // +2112
#define LDS_WS  (LDS_WM + NWAVES * 16)
#define LDS_MR  (LDS_WS + NWAVES * 16)
#define LDS_ZR  (LDS_MR + 16)
#define LDS_FLOATS (LDS_ZR + 16)                    // 35744 floats = 142976 B

typedef float v2f_t __attribute__((ext_vector_type(2)));
typedef float v8f_t __attribute__((ext_vector_type(8)));

// -----------------------------------------------------------------------------
// Panel staging helpers: each thread owns 32 floats (16 float2) of the
// 64(f) x 128(col) A_W_2 panel.  Loads go to registers first so the global
// latency overlaps the WMMA loop of the previous panel (software pipeline);
// stores write the panel TRANSPOSED (Pn[col][f], pitch 66) so each WMMA
// B-fragment is a single ds_load_2addr_b64.
// -----------------------------------------------------------------------------
__device__ inline void load_panel_regs(float2 (&buf)[16], const float* __restrict__ A2,
                                       int f0, int fr, int cg, int n0, bool full) {
  const float* src = A2 + (size_t)(f0 + fr) * ORIG;
  if (full) {
    #pragma unroll
    for (int i = 0; i < 16; ++i)
      buf[i] = *(const float2*)(src + n0 + cg + i * 2);   // 8B-aligned
  } else {                                                // tail chunk: clamp
    #pragma unroll
    for (int i = 0; i < 16; ++i) {
      int c0 = n0 + cg + i * 2;
      int c1 = c0 + 1;
      if (c0 >= ORIG) c0 = ORIG - 1;
      if (c1 >= ORIG) c1 = ORIG - 1;
      buf[i].x = src[c0];
      buf[i].y = src[c1];
    }
  }
}

__device__ inline void store_panel_regs(float* Pn, const float2 (&buf)[16],
                                        int fr, int cg) {
  #pragma unroll
  for (int i = 0; i < 16; ++i) {
    const int cc = cg + i * 2;
    Pn[(cc + 0) * PNPITCH + fr] = buf[i].x;               // transposed store
    Pn[(cc + 1) * PNPITCH + fr] = buf[i].y;
  }
}

// -----------------------------------------------------------------------------
// S-tile compute: this wave's 16-column slice of chunk n0:
//   S[16x16] = Q_blk[16x1024] x A_W_2[1024, cols]   (fp32 WMMA, K=4)
// Double-buffered panels: store panel p+1 / prefetch panel p+2 while the
// 16 WMMAs of panel p execute; ONE barrier per panel.
//
// WMMA f32 16x16x4 fragment layout (CDNA5 ISA 7.12.2):
//   A (16x4):  lane<16 -> M=lane, regs {K0,K1}; lane>=16 -> M=lane-16, {K2,K3}
//   B (4x16):  lane<16 -> N=lane, regs {K0,K1}; lane>=16 -> N=lane-16, {K2,K3}
//   C (16x16): VGPR r: lane<16 -> (M=r, N=lane); lane>=16 -> (M=r+8, N=lane-16)
// -----------------------------------------------------------------------------
__device__ inline v8f_t compute_S_tile(const float* __restrict__ A2,
                                       const float* Qs, float* PnA, float* PnB,
                                       int n0, bool full,
                                       int wave, int half, int l16) {
  v8f_t c = {0.f, 0.f, 0.f, 0.f, 0.f, 0.f, 0.f, 0.f};
  const int tid = threadIdx.x;
  const int fr = tid >> 2;          // 0..63 : f-row within panel
  const int cg = (tid & 3) * 32;    // column group of 32
  float2 buf[16];

  load_panel_regs(buf, A2, 0, fr, cg, n0, full);          // prologue: panel 0
  __syncthreads();                                        // prior Pn users done
  store_panel_regs(PnA, buf, fr, cg);
  load_panel_regs(buf, A2, KF, fr, cg, n0, full);         // prefetch panel 1
  __syncthreads();                                        // PnA ready

  for (int p = 0; p < NPANEL; ++p) {
    float* cur = (p & 1) ? PnB : PnA;
    float* nxt = (p & 1) ? PnA : PnB;
    if (p + 1 < NPANEL) {
      store_panel_regs(nxt, buf, fr, cg);                 // waits on loads
      if (p + 2 < NPANEL)
        load_panel_regs(buf, A2, (p + 2) * KF, fr, cg, n0, full);  // in flight
    }
    const float* qrow = Qs + l16 * QPITCH + p * KF + 2 * half;
    const float* prow = cur + (wave * 16 + l16) * PNPITCH + 2 * half;
    #pragma unroll
    for (int ks = 0; ks < KF; ks += 4) {
      v2f_t a = *(const v2f_t*)(qrow + ks);               // ds_load_2addr_b64
      v2f_t b = *(const v2f_t*)(prow + ks);               // ds_load_2addr_b64
      c = __builtin_amdgcn_wmma_f32_16x16x4_f32(false, a, false, b,
                                                (short)0, c, false, false);
    }
    __syncthreads();   // cur reads done everywhere; nxt stores visible
  }
  return c;
}

// -----------------------------------------------------------------------------
// Phase 1: fused  new_emb = softmax(A_W_1 @ A_W_2) @ W_original
// One workgroup (8 waves) per 16 NEW rows; two passes over ORIG:
//   pass 1: online row-max / row-sumexp (per-wave, merged via LDS)
//   pass 2: recompute logits, P = exp(s-m)/Z, accumulate P @ W_original
// A_W_2 (82 MB) + W_original (61 MB) both fit in the 192 MB L2, so after the
// first sweep the stream is L2-resident and the kernel is fp32-matrix bound.
// -----------------------------------------------------------------------------
__global__ __launch_bounds__(256, 1)
void new_emb_kernel(const float* __restrict__ A1, const float* __restrict__ A2,
                    const float* __restrict__ W, float* __restrict__ new_emb) {
  extern __shared__ float smem[];
  float* Qs   = smem + LDS_Q;
  float* PnA  = smem + LDS_PNA;
  float* PnB  = smem + LDS_PNB;
  float* Pb   = smem + LDS_PB;
  float* Wm   = smem + LDS_WM;
  float* Wsx  = smem + LDS_WS;
  float* Mrow = smem + LDS_MR;
  float* Zrow = smem + LDS_ZR;

  const int tid  = threadIdx.x;
  const int wave = tid >> 5;
  const int lane = tid & 31;
  const int half = lane >> 4;
  const int l16  = lane & 15;
  const int row0 = blockIdx.x * BM;

  // ---- stage Q block (16 x 1024) into LDS, rows clamped at the NEW tail ----
  {
    const int r = tid >> 4;                 // 0..15
    const int cb = (tid & 15) * 64;         // 64 floats per thread
    int gr = row0 + r; if (gr >= NEWROWS) gr = NEWROWS - 1;
    const float* src = A1 + (size_t)gr * FACT + cb;
    float* dst = Qs + r * QPITCH + cb;
    #pragma unroll
    for (int i = 0; i < 16; ++i)
      *(float4*)(dst + i * 4) = *(const float4*)(src + i * 4);
  }

  // ----------------------------- pass 1 ------------------------------------
  float run_m[8], run_s[8];
  #pragma unroll
  for (int r = 0; r < 8; ++r) { run_m[r] = -__builtin_inff(); run_s[r] = 0.f; }

  for (int ch = 0; ch < NCHUNK; ++ch) {
    const int n0 = ch * BN;
    const bool full = (n0 + BN) <= ORIG;
    if (n0 + BN < ORIG)     // warm L2 for next chunk's first panel
      __builtin_prefetch(A2 + (size_t)(tid >> 2) * ORIG + n0 + BN + (tid & 3) * 32, 0, 0);
    v8f_t c = compute_S_tile(A2, Qs, PnA, PnB, n0, full, wave, half, l16);
    const bool cvalid = (n0 + wave * 16 + l16) < ORIG;
    #pragma unroll
    for (int r = 0; r < 8; ++r) {
      float s = cvalid ? c[r] : -__builtin_inff();
      float mx = s;
      #pragma unroll
      for (int m = 1; m < 16; m <<= 1) mx = fmaxf(mx, __shfl_xor(mx, m, 16));
      const float nm = fmaxf(run_m[r], mx);
      float e = cvalid ? __expf(s - nm) : 0.f;
      #pragma unroll
      for (int m = 1; m < 16; m <<= 1) e += __shfl_xor(e, m, 16);
      run_s[r] = (nm == run_m[r]) ? (run_s[r] + e)
                                  : (run_s[r] * __expf(run_m[r] - nm) + e);
      run_m[r] = nm;
    }
  }

  // merge per-wave (m, sumexp) across the 8 waves
  if (l16 == 0) {
    #pragma unroll
    for (int r = 0; r < 8; ++r) {
      Wm [wave * 16 + r + 8 * half] = run_m[r];
      Wsx[wave * 16 + r + 8 * half] = run_s[r];
    }
  }
  __syncthreads();
  if (tid < 16) {
    float m = -__builtin_inff();
    for (int w = 0; w < NWAVES; ++w) m = fmaxf(m, Wm[w * 16 + tid]);
    float z = 0.f;
    for (int w = 0; w < NWAVES; ++w) z += Wsx[w * 16 + tid] * __expf(Wm[w * 16 + tid] - m);
    Mrow[tid] = m;
    Zrow[tid] = z;
  }
  __syncthreads();

  float mr[8], zr[8];
  #pragma unroll
  for (int r = 0; r < 8; ++r) {
    mr[r] = Mrow[r + 8 * half];
    zr[r] = 1.0f / Zrow[r + 8 * half];
  }

  // ----------------------------- pass 2 ------------------------------------
  v8f_t acc[6];
  #pragma unroll
  for (int j = 0; j < 6; ++j) acc[j] = (v8f_t){0.f,0.f,0.f,0.f,0.f,0.f,0.f,0.f};

  for (int ch = 0; ch < NCHUNK; ++ch) {
    const int n0 = ch * BN;
    const bool full = (n0 + BN) <= ORIG;
    if (n0 + BN < ORIG)
      __builtin_prefetch(A2 + (size_t)(tid >> 2) * ORIG + n0 + BN + (tid & 3) * 32, 0, 0);
    v8f_t c = compute_S_tile(A2, Qs, PnA, PnB, n0, full, wave, half, l16);

    // P = exp(s - m)/Z  (masked at the ORIG tail), shared via LDS
    const int colL = wave * 16 + l16;
    const bool cvalid = (n0 + colL) < ORIG;
    #pragma unroll
    for (int r = 0; r < 8; ++r) {
      const float p = cvalid ? __expf(c[r] - mr[r]) * zr[r] : 0.f;
      Pb[(r + 8 * half) * PPITCH + colL] = p;
    }
    __syncthreads();

    // acc += P[16 x 128] @ W_original[chunk, dim-slice]; wave owns 96 DIM cols
    const int d0 = wave * 96;
    for (int ks = 0; ks < BN; ks += 4) {
      v2f_t a = *(const v2f_t*)(Pb + l16 * PPITCH + ks + 2 * half);
      const int k0 = n0 + ks + 2 * half;
      const int kr0 = (k0     < ORIG) ? k0     : ORIG - 1;   // P==0 there anyway
      const int kr1 = (k0 + 1 < ORIG) ? k0 + 1 : ORIG - 1;
      const float* wb0 = W + (size_t)kr0 * DIM;
      const float* wb1 = W + (size_t)kr1 * DIM;
      #pragma unroll
      for (int j = 0; j < 6; ++j) {
        const int dc = d0 + j * 16 + l16;
        v2f_t b; b.x = wb0[dc]; b.y = wb1[dc];               // L2-resident reads
        acc[j] = __builtin_amdgcn_wmma_f32_16x16x4_f32(false, a, false, b,
                                                       (short)0, acc[j], false, false);
      }
    }
    __syncthreads();          // Pb consumed before next chunk overwrites it
  }

  // ---- store this block's rows of new_emb ----
  #pragma unroll
  for (int j = 0; j < 6; ++j) {
    #pragma unroll
    for (int r = 0; r < 8; ++r) {
      const int row = row0 + r + 8 * half;
      if (row < NEWROWS)
        new_emb[(size_t)row * DIM + wave * 96 + j * 16 + l16] = acc[j][r];
    }
  }
}

// -----------------------------------------------------------------------------
// Phase 2: gathered dot products + log-sigmoid losses. One block per batch
// element; fixed-order reductions (deterministic, no float atomics).
// -----------------------------------------------------------------------------
__device__ inline float softplusf(float y) {            // log(1 + e^y), stable
  return fmaxf(y, 0.0f) + log1pf(__expf(-fabsf(y)));
}

__global__ __launch_bounds__(256)
void loss_kernel(const float* __restrict__ W, const float* __restrict__ newE,
                 const int* __restrict__ in_ids, const int* __restrict__ pos_ids,
                 const int* __restrict__ neg_ids, float* __restrict__ partials) {
  const int b = blockIdx.x;
  const int tid = threadIdx.x;

  const int iid = in_ids[b];
  const int pid = pos_ids[b];
  const float* ein  = (iid < ORIG) ? W + (size_t)iid * DIM : newE + (size_t)(iid - ORIG) * DIM;
  const float* epos = (pid < ORIG) ? W + (size_t)pid * DIM : newE + (size_t)(pid - ORIG) * DIM;
  const float* en[KNEG];
  #pragma unroll
  for (int k = 0; k < KNEG; ++k) {
    const int nid = neg_ids[b * KNEG + k];
    en[k] = (nid < ORIG) ? W + (size_t)nid * DIM : newE + (size_t)(nid - ORIG) * DIM;
  }

  float accP = 0.f, accN = 0.f;                 // only the sum over k is needed
  for (int c = tid; c < DIM; c += 256) {
    const float vi = ein[c];
    accP += epos[c] * vi;
    float ns = 0.f;
    #pragma unroll
    for (int k = 0; k < KNEG; ++k) ns += en[k][c];
    accN += ns * vi;
  }
  #pragma unroll
  for (int m = 1; m < 32; m <<= 1) {
    accP += __shfl_xor(accP, m, 32);
    accN += __shfl_xor(accN, m, 32);
  }
  __shared__ float sP[8], sN[8];
  if ((tid & 31) == 0) { sP[tid >> 5] = accP; sN[tid >> 5] = accN; }
  __syncthreads();
  if (tid == 0) {
    float P = 0.f, N = 0.f;
    #pragma unroll
    for (int w = 0; w < 8; ++w) { P += sP[w]; N += sN[w]; }
    partials[b] = softplusf(-P) + softplusf(N);   // -logsig(P) + -logsig(-N)
  }
}

__global__ __launch_bounds__(256)
void reduce_kernel(const float* __restrict__ partials, float* __restrict__ out) {
  __shared__ float sh[256];
  const int tid = threadIdx.x;
  float s = 0.f;
  for (int i = tid; i < BATCH; i += 256) s += partials[i];
  sh[tid] = s;
  __syncthreads();
  for (int off = 128; off > 0; off >>= 1) {
    if (tid < off) sh[tid] += sh[tid + off];
    __syncthreads();
  }
  if (tid == 0) out[0] = sh[0] / (float)BATCH;
}

// -----------------------------------------------------------------------------
extern "C" void kernel_launch(void* const* d_in, const int* in_sizes, int n_in,
                              void* d_out, int out_size, void* d_ws, size_t ws_size,
                              hipStream_t stream) {
  const float* W   = (const float*)d_in[0];   // [ORIG, DIM]
  const float* A1  = (const float*)d_in[1];   // [NEW, FACT]
  const float* A2  = (const float*)d_in[2];   // [FACT, ORIG]
  const int* in_ids  = (const int*)d_in[3];   // [B]
  const int* pos_ids = (const int*)d_in[4];   // [B]
  const int* neg_ids = (const int*)d_in[5];   // [B, K]
  float* out = (float*)d_out;

  float* new_emb  = (float*)d_ws;                                   // 116.6 MB
  float* partials = (float*)((char*)d_ws + (size_t)NEWROWS * DIM * sizeof(float));

  new_emb_kernel<<<GRID1, 256, LDS_FLOATS * sizeof(float), stream>>>(A1, A2, W, new_emb);
  loss_kernel<<<BATCH, 256, 0, stream>>>(W, new_emb, in_ids, pos_ids, neg_ids, partials);
  reduce_kernel<<<1, 256, 0, stream>>>(partials, out);
}